// GraphDenoiserMasked_90709709291551
// MI455X (gfx1250) — compile-verified
//
#include <hip/hip_runtime.h>
#include <math.h>

#define CC   256
#define HEADS 8
#define HD   32
#define NB   16
#define NT   120
#define NJ   25
#define NTOK (NB*NT*NJ)   /* 48000 */

typedef __attribute__((ext_vector_type(2))) float v2f;
typedef __attribute__((ext_vector_type(8))) float v8f;

#if __has_builtin(__builtin_amdgcn_global_load_async_to_lds_b32)
#define ASYNC_W 1
typedef __attribute__((address_space(1))) int as1_int;
typedef __attribute__((address_space(3))) int as3_int;
#else
#define ASYNC_W 0
#endif

__device__ __forceinline__ float gelu_f(float x) {
  return 0.5f * x * (1.0f + erff(x * 0.7071067811865476f));
}

// Stage one W element into LDS fragment slot (async-to-LDS when available).
__device__ __forceinline__ void stage_w(const float* __restrict__ gsrc, float* ldst) {
#if ASYNC_W
  __builtin_amdgcn_global_load_async_to_lds_b32(
      (as1_int*)gsrc, (as3_int*)ldst, 0, 0);
#else
  *ldst = *gsrc;
#endif
}

__device__ __forceinline__ void stage_fence() {
#if ASYNC_W
  asm volatile("s_wait_asynccnt 0x0" ::: "memory");
#endif
  __syncthreads();
}

// ---------------------------------------------------------------------------
// Generic fp32 WMMA GEMM: out[R,N] = act( X[R,K] @ W[K,N] + bias + res + bcasts )
// Block = 256 threads (8 waves). Each wave -> one 16x32 tile (two 16x16 WMMA
// accumulators sharing each A fragment -> halves A-side global traffic).
// Block tile: 128 rows x 32 cols. Requires R % 128 == 0, N % 32 == 0, K % 256 == 0.
// K is staged through LDS in 256-deep chunks (32 KB static LDS) in B-fragment
// order: lane<16 holds (W[4g+0,n], W[4g+1,n]); lane>=16 holds (W[4g+2,n], W[4g+3,n]).
// ---------------------------------------------------------------------------
__global__ void __launch_bounds__(256)
wmma_gemm_kernel(const float* __restrict__ X, const float* __restrict__ W,
                 const float* __restrict__ bias, const float* __restrict__ res,
                 const float* __restrict__ bA, const float* __restrict__ bB,
                 float* __restrict__ out, int K, int N, int actGelu,
                 int bAdiv, int bBdiv)
{
  __shared__ float s_w[2][64 * 64];         // two 256x16 tiles, fragment order
  const int tid  = threadIdx.x;
  const int lane = tid & 31;
  const int wave = tid >> 5;
  const int n0 = blockIdx.x * 32;
  const int m0 = blockIdx.y * 128 + wave * 16;

  v8f c0 = {0.f, 0.f, 0.f, 0.f, 0.f, 0.f, 0.f, 0.f};
  v8f c1 = {0.f, 0.f, 0.f, 0.f, 0.f, 0.f, 0.f, 0.f};

  const float* xrow = X + (size_t)(m0 + (lane & 15)) * K + ((lane >> 4) << 1);

  for (int k0 = 0; k0 < K; k0 += 256) {
    __syncthreads();                        // previous chunk fully consumed
    // cooperative W-chunk staging into fragment layout (async-to-LDS if present)
    for (int e = tid; e < 256 * 32; e += 256) {
      int kl = e >> 5;                      // 0..255 within chunk
      int n  = e & 31;
      int g  = kl >> 2;
      int rr = kl & 3;
      int l  = ((rr >> 1) << 4) | (n & 15);
      stage_w(W + (size_t)(k0 + kl) * N + n0 + n,
              &s_w[n >> 4][g * 64 + l * 2 + (rr & 1)]);
    }
    stage_fence();

    const float* xp = xrow + k0;
#pragma unroll 4
    for (int g = 0; g < 64; ++g) {
      v2f a  = *(const v2f*)(xp + g * 4);                  // global_load_b64
      v2f b0 = *(const v2f*)(&s_w[0][g * 64 + lane * 2]);  // ds_load_b64
      v2f b1 = *(const v2f*)(&s_w[1][g * 64 + lane * 2]);
      c0 = __builtin_amdgcn_wmma_f32_16x16x4_f32(false, a, false, b0,
                                                 (short)0, c0, false, false);
      c1 = __builtin_amdgcn_wmma_f32_16x16x4_f32(false, a, false, b1,
                                                 (short)0, c1, false, false);
    }
  }

  // Epilogue. C/D layout: vgpr i -> row m0 + i + (lane>=16 ? 8 : 0), col base + (lane&15)
  const int colL = lane & 15;
  const int rb   = m0 + ((lane >> 4) << 3);
  auto epi = [&](const v8f& c, int colBase) {
    const int col = colBase + colL;
    const float bv = bias ? bias[col] : 0.f;
#pragma unroll
    for (int i = 0; i < 8; ++i) {
      int row = rb + i;
      float x = c[i] + bv;
      if (bA)  x += bA[(size_t)(row / bAdiv) * CC + col];  // broadcasts only when N==256
      if (bB)  x += bB[(size_t)(row / bBdiv) * CC + col];
      if (res) x += res[(size_t)row * N + col];
      if (actGelu) x = gelu_f(x);
      out[(size_t)row * N + col] = x;
    }
  };
  epi(c0, n0);
  epi(c1, n0 + 16);
}

// ---------------------------------------------------------------------------
// Temporal conv (k=3, pad 1, over T) as 3 accumulated GEMMs.
// y[r,:] = sum_tap X[r+(tap-1)*J, :] @ Wtap   (Wtap[cin][cout] = Wc[cout*768+cin*3+tap])
// Out-of-range t handled by zeroing A fragments per lane (WMMA keeps full EXEC).
// ---------------------------------------------------------------------------
__global__ void __launch_bounds__(256)
wmma_conv_kernel(const float* __restrict__ X, const float* __restrict__ Wc,
                 const float* __restrict__ bias, const float* __restrict__ res,
                 float* __restrict__ out, int actGelu)
{
  __shared__ float s_w[2][64 * 64];
  const int tid  = threadIdx.x;
  const int lane = tid & 31;
  const int wave = tid >> 5;
  const int n0 = blockIdx.x * 32;
  const int m0 = blockIdx.y * 128 + wave * 16;

  const int rowA = m0 + (lane & 15);
  const int tt   = (rowA / NJ) % NT;

  v8f c0 = {0.f, 0.f, 0.f, 0.f, 0.f, 0.f, 0.f, 0.f};
  v8f c1 = {0.f, 0.f, 0.f, 0.f, 0.f, 0.f, 0.f, 0.f};

  for (int tap = 0; tap < 3; ++tap) {
    __syncthreads();
    for (int e = tid; e < 256 * 32; e += 256) {
      int kl = e >> 5;
      int n  = e & 31;
      int g  = kl >> 2;
      int rr = kl & 3;
      int l  = ((rr >> 1) << 4) | (n & 15);
      stage_w(Wc + (size_t)(n0 + n) * 768 + kl * 3 + tap,
              &s_w[n >> 4][g * 64 + l * 2 + (rr & 1)]);
    }
    stage_fence();

    const int t2 = tt + tap - 1;
    const bool valid = (t2 >= 0) && (t2 < NT);
    const float* xp = X + (size_t)(rowA + (tap - 1) * NJ) * CC + ((lane >> 4) << 1);
#pragma unroll 4
    for (int g = 0; g < 64; ++g) {
      v2f a = {0.f, 0.f};
      if (valid) a = *(const v2f*)(xp + g * 4);
      v2f b0 = *(const v2f*)(&s_w[0][g * 64 + lane * 2]);
      v2f b1 = *(const v2f*)(&s_w[1][g * 64 + lane * 2]);
      c0 = __builtin_amdgcn_wmma_f32_16x16x4_f32(false, a, false, b0,
                                                 (short)0, c0, false, false);
      c1 = __builtin_amdgcn_wmma_f32_16x16x4_f32(false, a, false, b1,
                                                 (short)0, c1, false, false);
    }
  }

  const int colL = lane & 15;
  const int rb   = m0 + ((lane >> 4) << 3);
  auto epi = [&](const v8f& c, int colBase) {
    const int col = colBase + colL;
    const float bv = bias[col];
#pragma unroll
    for (int i = 0; i < 8; ++i) {
      int row = rb + i;
      float x = c[i] + bv;
      if (res) x += res[(size_t)row * CC + col];
      if (actGelu) x = gelu_f(x);
      out[(size_t)row * CC + col] = x;
    }
  };
  epi(c0, n0);
  epi(c1, n0 + 16);
}

// ---------------------------------------------------------------------------
// LayerNorm: one wave per row of 256, wave32 shfl reductions.
// ---------------------------------------------------------------------------
__global__ void __launch_bounds__(256)
ln_kernel(const float* __restrict__ x, const float* __restrict__ g,
          const float* __restrict__ b, float* __restrict__ y, int rows)
{
  const int lane = threadIdx.x & 31;
  const int row  = blockIdx.x * 8 + (threadIdx.x >> 5);
  if (row >= rows) return;
  const float* xr = x + (size_t)row * CC;
  float v[8];
  float s = 0.f;
#pragma unroll
  for (int i = 0; i < 8; ++i) { v[i] = xr[lane + 32 * i]; s += v[i]; }
#pragma unroll
  for (int o = 16; o > 0; o >>= 1) s += __shfl_xor(s, o);
  const float mean = s * (1.0f / 256.0f);
  float s2 = 0.f;
#pragma unroll
  for (int i = 0; i < 8; ++i) { float d = v[i] - mean; s2 += d * d; }
#pragma unroll
  for (int o = 16; o > 0; o >>= 1) s2 += __shfl_xor(s2, o);
  const float rstd = rsqrtf(s2 * (1.0f / 256.0f) + 1e-5f);
  float* yr = y + (size_t)row * CC;
#pragma unroll
  for (int i = 0; i < 8; ++i) {
    int cc = lane + 32 * i;
    yr[cc] = (v[i] - mean) * rstd * g[cc] + b[cc];
  }
}

// ---------------------------------------------------------------------------
// Attention over the J=25 graph, one wave per (b,t,head). ~1.4% of FLOPs.
// ---------------------------------------------------------------------------
__global__ void __launch_bounds__(32)
attn_kernel(const float* __restrict__ q, const float* __restrict__ k,
            const float* __restrict__ v, const int* __restrict__ adj,
            float* __restrict__ att)
{
  const int bt = blockIdx.x >> 3;    // b*T + t
  const int h  = blockIdx.x & 7;
  const int lane = threadIdx.x;
  __shared__ float qs[NJ][HD], ks[NJ][HD], vs[NJ][HD];
  const size_t base = (size_t)bt * NJ * CC + h * HD;
  for (int j = 0; j < NJ; ++j) {
    qs[j][lane] = q[base + (size_t)j * CC + lane];
    ks[j][lane] = k[base + (size_t)j * CC + lane];
    vs[j][lane] = v[base + (size_t)j * CC + lane];
  }
  __syncthreads();
  if (lane < NJ) {
    const int i = lane;
    float s[NJ];
    float mx = -1e30f;
    for (int j = 0; j < NJ; ++j) {
      float acc = 0.f;
#pragma unroll
      for (int d = 0; d < HD; ++d) acc += qs[i][d] * ks[j][d];
      acc *= 0.17677669529663687f;                 // 1/sqrt(32)
      bool ok = (i == j) || (adj[i * NJ + j] != 0);
      s[j] = ok ? acc : -10000.0f;
      mx = fmaxf(mx, s[j]);
    }
    float sum = 0.f;
    for (int j = 0; j < NJ; ++j) { s[j] = expf(s[j] - mx); sum += s[j]; }
    const float inv = 1.0f / sum;
    for (int d = 0; d < HD; ++d) {
      float acc = 0.f;
      for (int j = 0; j < NJ; ++j) acc += s[j] * vs[j][d];
      att[base + (size_t)i * CC + d] = acc * inv;
    }
  }
}

// ---------------------------------------------------------------------------
// Timestep embedding + 2-layer time MLP (tiny: 16 rows). One block.
// ---------------------------------------------------------------------------
__global__ void __launch_bounds__(256)
time_kernel(const int* __restrict__ t, const float* __restrict__ W1,
            const float* __restrict__ b1, const float* __restrict__ W2,
            const float* __restrict__ b2, float* __restrict__ te)
{
  __shared__ float emb[NB * CC];
  __shared__ float mid[NB * CC];
  const int tid = threadIdx.x;
  const int fi = tid & 127;
  const float freq = expf((float)fi * (-9.210340371976184f / 127.0f)); // -ln(1e4)/(half-1)
  for (int bb = 0; bb < NB; ++bb) {
    float ang = (float)t[bb] * freq;
    emb[bb * CC + tid] = (tid < 128) ? sinf(ang) : cosf(ang);
  }
  __syncthreads();
  for (int bb = 0; bb < NB; ++bb) {
    float acc = b1[tid];
    for (int kk = 0; kk < CC; ++kk) acc += emb[bb * CC + kk] * W1[kk * CC + tid];
    mid[bb * CC + tid] = gelu_f(acc);
  }
  __syncthreads();
  for (int bb = 0; bb < NB; ++bb) {
    float acc = b2[tid];
    for (int kk = 0; kk < CC; ++kk) acc += mid[bb * CC + kk] * W2[kk * CC + tid];
    te[bb * CC + tid] = acc;
  }
}

// ---------------------------------------------------------------------------
extern "C" void kernel_launch(void* const* d_in, const int* in_sizes, int n_in,
                              void* d_out, int out_size, void* d_ws, size_t ws_size,
                              hipStream_t stream) {
  (void)in_sizes; (void)n_in; (void)out_size; (void)ws_size;

  const float* z_t = (const float*)d_in[0];
  const int*   t   = (const int*)d_in[1];
  const int*   adj = (const int*)d_in[2];
  const float* h   = (const float*)d_in[3];

  int p = 4;
  auto nxt = [&]() { return (const float*)d_in[p++]; };
  const float *inW = nxt(), *inB = nxt();
  const float *t1W = nxt(), *t1B = nxt();
  const float *t2W = nxt(), *t2B = nxt();
  const float *cpW = nxt(), *cpB = nxt();
  struct Layer {
    const float *n1g,*n1b,*n2g,*n2b,*qW,*qB,*kW,*kB,*vW,*vB,*oW,*oB,*f1W,*f1B,*f2W,*f2B;
  } L[4];
  for (int l = 0; l < 4; ++l) {
    L[l].n1g = nxt(); L[l].n1b = nxt();
    L[l].n2g = nxt(); L[l].n2b = nxt();
    L[l].qW  = nxt(); L[l].qB  = nxt();
    L[l].kW  = nxt(); L[l].kB  = nxt();
    L[l].vW  = nxt(); L[l].vB  = nxt();
    L[l].oW  = nxt(); L[l].oB  = nxt();
    L[l].f1W = nxt(); L[l].f1B = nxt();
    L[l].f2W = nxt(); L[l].f2B = nxt();
  }
  const float *c1W = nxt(), *c1B = nxt();
  const float *c2W = nxt(), *c2B = nxt();
  const float *onG = nxt(), *onB = nxt();
  const float *opW = nxt(), *opB = nxt();

  // workspace layout (floats)
  float* ws   = (float*)d_ws;
  float* te   = ws;                                  // 16*256
  float* ch   = te + (size_t)NB * CC;                // 1920*256
  float* xn   = ch + (size_t)NB * NT * CC;           // 48000*256
  float* pool = xn + (size_t)NTOK * CC;              // 48000*1024 (q|k|v|att / ffn)
  float* qb   = pool;
  float* kb   = pool + 1 * (size_t)NTOK * CC;
  float* vb   = pool + 2 * (size_t)NTOK * CC;
  float* ab   = pool + 3 * (size_t)NTOK * CC;
  float* hid  = pool + 4 * (size_t)NTOK * CC;        // 48000*256
  float* outp = (float*)d_out;

  const dim3 blk(256);
  const dim3 gFull(CC / 32, NTOK / 128);         // 256-wide GEMMs over 48000 rows
  const dim3 gFfn1(1024 / 32, NTOK / 128);       // 1024-wide output
  const dim3 gCond(CC / 32, (NB * NT) / 128);    // 1920 rows

  // time MLP -> te[16,256]
  time_kernel<<<1, 256, 0, stream>>>(t, t1W, t1B, t2W, t2B, te);
  // cond proj: ch = h @ cpW + cpB
  wmma_gemm_kernel<<<gCond, blk, 0, stream>>>(h, cpW, cpB, nullptr, nullptr, nullptr,
                                              ch, 256, 256, 0, 1, 1);
  // hid = z_t @ inW + inB + te[b] + ch[b,t]
  wmma_gemm_kernel<<<gFull, blk, 0, stream>>>(z_t, inW, inB, nullptr, te, ch,
                                              hid, 256, 256, 0, NT * NJ, NJ);

  for (int l = 0; l < 4; ++l) {
    ln_kernel<<<NTOK / 8, 256, 0, stream>>>(hid, L[l].n1g, L[l].n1b, xn, NTOK);
    wmma_gemm_kernel<<<gFull, blk, 0, stream>>>(xn, L[l].qW, L[l].qB, nullptr, nullptr, nullptr,
                                                qb, 256, 256, 0, 1, 1);
    wmma_gemm_kernel<<<gFull, blk, 0, stream>>>(xn, L[l].kW, L[l].kB, nullptr, nullptr, nullptr,
                                                kb, 256, 256, 0, 1, 1);
    wmma_gemm_kernel<<<gFull, blk, 0, stream>>>(xn, L[l].vW, L[l].vB, nullptr, nullptr, nullptr,
                                                vb, 256, 256, 0, 1, 1);
    attn_kernel<<<NB * NT * HEADS, 32, 0, stream>>>(qb, kb, vb, adj, ab);
    // hid = hid + att @ oW + oB
    wmma_gemm_kernel<<<gFull, blk, 0, stream>>>(ab, L[l].oW, L[l].oB, hid, nullptr, nullptr,
                                                hid, 256, 256, 0, 1, 1);
    ln_kernel<<<NTOK / 8, 256, 0, stream>>>(hid, L[l].n2g, L[l].n2b, xn, NTOK);
    // ffn: pool = gelu(xn @ f1W + f1B) ; hid += pool @ f2W + f2B
    wmma_gemm_kernel<<<gFfn1, blk, 0, stream>>>(xn, L[l].f1W, L[l].f1B, nullptr, nullptr, nullptr,
                                                pool, 256, 1024, 1, 1, 1);
    wmma_gemm_kernel<<<gFull, blk, 0, stream>>>(pool, L[l].f2W, L[l].f2B, hid, nullptr, nullptr,
                                                hid, 1024, 256, 0, 1, 1);
  }

  // temporal: hid += conv2(gelu(conv1(hid)))
  wmma_conv_kernel<<<gFull, blk, 0, stream>>>(hid, c1W, c1B, nullptr, xn, 1);
  wmma_conv_kernel<<<gFull, blk, 0, stream>>>(xn, c2W, c2B, hid, hid, 0);

  // out = LN(hid) @ opW + opB
  ln_kernel<<<NTOK / 8, 256, 0, stream>>>(hid, onG, onB, xn, NTOK);
  wmma_gemm_kernel<<<gFull, blk, 0, stream>>>(xn, opW, opB, nullptr, nullptr, nullptr,
                                              outp, 256, 256, 0, 1, 1);
}